// StructuralCoherenceLoss_54666343744093
// MI455X (gfx1250) — compile-verified
//
#include <hip/hip_runtime.h>
#include <hip/hip_bf16.h>

typedef __attribute__((ext_vector_type(2))) float v2f;
typedef __attribute__((ext_vector_type(8))) float v8f;

__device__ __forceinline__ bool is_air(int t) {
    return (t == 102) | (t == 576) | (t == 3352);
}

// Exact wave32 sum via V_WMMA_F32_16X16X4_F32.
// A holds one partial per lane (a.y = 0), B = ones(4x16), C = 0.
// D[m,n] = rowsum_m (replicated over n). Per-lane t = sum of 8 D VGPRs:
// lanes 0..15 hold sum over rows 0..7, lanes 16..31 over rows 8..15.
// total = t(lane0) + t(lane16). Layout-agnostic as long as A slots are a
// bijection onto matrix entries (they are, per ISA 16x4 f32 A layout).
__device__ __forceinline__ float wave_sum_wmma(float v) {
    v2f a; a.x = v;    a.y = 0.0f;
    v2f b; b.x = 1.0f; b.y = 1.0f;
    v8f c = {};
    c = __builtin_amdgcn_wmma_f32_16x16x4_f32(false, a, false, b, (short)0, c,
                                              false, false);
    float t = c[0] + c[1] + c[2] + c[3] + c[4] + c[5] + c[6] + c[7];
    return __shfl(t, 0, 32) + __shfl(t, 16, 32);
}

__global__ void k_init(double* __restrict__ ws) {
    if (threadIdx.x < 7) ws[threadIdx.x] = 0.0;
}

// ws[0]=isolated, ws[1]=non_air, ws[2]=unsupported
__global__ void k_struct(const int* __restrict__ s, double* __restrict__ ws) {
    const int VOX = 64 * 32 * 32 * 32;
    int tid = blockIdx.x * blockDim.x + threadIdx.x;
    int stride = gridDim.x * blockDim.x;
    float iso = 0.0f, nac = 0.0f, uns = 0.0f;
    for (int v = tid; v < VOX; v += stride) {
        int t = s[v];
        if (!is_air(t)) {
            int z = v & 31, y = (v >> 5) & 31, x = (v >> 10) & 31;
            int nb = 0;
            if (x > 0)  nb += !is_air(s[v - 1024]);
            if (x < 31) nb += !is_air(s[v + 1024]);
            if (y > 0)  nb += !is_air(s[v - 32]);
            if (y < 31) nb += !is_air(s[v + 32]);
            if (z > 0)  nb += !is_air(s[v - 1]);
            if (z < 31) nb += !is_air(s[v + 1]);
            nac += 1.0f;
            if (nb == 0) iso += 1.0f;
            if (y > 0 && is_air(s[v - 32])) uns += 1.0f;
        }
    }
    float wi = wave_sum_wmma(iso);
    float wn = wave_sum_wmma(nac);
    float wu = wave_sum_wmma(uns);
    __shared__ double acc[3];
    if (threadIdx.x < 3) acc[threadIdx.x] = 0.0;
    __syncthreads();
    if ((threadIdx.x & 31) == 0) {
        atomicAdd(&acc[0], (double)wi);
        atomicAdd(&acc[1], (double)wn);
        atomicAdd(&acc[2], (double)wu);
    }
    __syncthreads();
    if (threadIdx.x == 0) {
        atomicAdd(&ws[0], acc[0]);
        atomicAdd(&ws[1], acc[1]);
        atomicAdd(&ws[2], acc[2]);
    }
}

// ws[3] = sum of per-patch entropies over all 32768 patches.
// 4 patches per 256-thread block; 64 threads per 4x4x4 patch.
__global__ void k_patch(const int* __restrict__ s, double* __restrict__ ws) {
    __shared__ int s_tok[256];
    __shared__ int s_na[256];
    int grp  = threadIdx.x >> 6;     // patch slot in block
    int lane = threadIdx.x & 63;     // element in patch (ix*16 + iy*4 + iz)
    int patch = blockIdx.x * 4 + grp;            // (px*64 + py*8 + pz) per batch
    int b   = patch >> 9;
    int rem = patch & 511;
    int px = rem >> 6, py = (rem >> 3) & 7, pz = rem & 7;
    int ix = lane >> 4, iy = (lane >> 2) & 3, iz = lane & 3;
    int X = px * 4 + ix, Y = py * 4 + iy, Z = pz * 4 + iz;
    int t = s[b * 32768 + X * 1024 + Y * 32 + Z];
    int na = !is_air(t);
    s_tok[threadIdx.x] = t;
    s_na[threadIdx.x] = na;
    __syncthreads();
    int base = grp << 6;
    int cnt = 0, ssum = 0;
    #pragma unroll 8
    for (int j = 0; j < 64; ++j) {
        cnt  += (s_tok[base + j] == t);  // air tokens never equal a non-air t
        ssum += s_na[base + j];
    }
    // Each non-air element contributes -(1/s)*log(c_t/s + 1e-10); summed over a
    // token with count c this is -(c/s)*log(c/s + 1e-10)  == reference entropy.
    float h = 0.0f;
    if (na && ssum > 0) {
        float fs = (float)ssum;
        h = -(1.0f / fs) * logf((float)cnt / fs + 1e-10f);
    }
    float wh = wave_sum_wmma(h);
    __shared__ double acc;
    if (threadIdx.x == 0) acc = 0.0;
    __syncthreads();
    if ((threadIdx.x & 31) == 0) atomicAdd(&acc, (double)wh);
    __syncthreads();
    if (threadIdx.x == 0) atomicAdd(&ws[3], acc);
}

// ws[4]=sum|dx|, ws[5]=sum|dy|, ws[6]=sum|dz|. One 32-float z-row per loop iter.
__global__ void k_smooth(const float* __restrict__ e, double* __restrict__ ws) {
    const int NROWS = 64 * 32 * 32 * 32;   // (B*C=2048) * 32x * 32y rows
    int tid = blockIdx.x * blockDim.x + threadIdx.x;
    int stride = gridDim.x * blockDim.x;
    float sx = 0.0f, sy = 0.0f, sz = 0.0f;
    for (int r = tid; r < NROWS; r += stride) {
        int y = r & 31, x = (r >> 5) & 31;
        size_t base = (size_t)r * 32;
        const float4* rp = (const float4*)(e + base);
        float4 row[8];
        #pragma unroll
        for (int i = 0; i < 8; ++i) row[i] = rp[i];
        float last = row[0].x;
        #pragma unroll
        for (int i = 0; i < 8; ++i) {
            float4 v = row[i];
            if (i) sz += fabsf(v.x - last);
            sz += fabsf(v.y - v.x) + fabsf(v.z - v.y) + fabsf(v.w - v.z);
            last = v.w;
        }
        if (y < 31) {
            const float4* np = (const float4*)(e + base + 32);
            #pragma unroll
            for (int i = 0; i < 8; ++i) {
                float4 n = np[i];
                sy += fabsf(n.x - row[i].x) + fabsf(n.y - row[i].y) +
                      fabsf(n.z - row[i].z) + fabsf(n.w - row[i].w);
            }
        }
        if (x < 31) {
            const float4* np = (const float4*)(e + base + 1024);
            #pragma unroll
            for (int i = 0; i < 8; ++i) {
                float4 n = np[i];
                sx += fabsf(n.x - row[i].x) + fabsf(n.y - row[i].y) +
                      fabsf(n.z - row[i].z) + fabsf(n.w - row[i].w);
            }
        }
    }
    float wx = wave_sum_wmma(sx);
    float wy = wave_sum_wmma(sy);
    float wz = wave_sum_wmma(sz);
    __shared__ double acc[3];
    if (threadIdx.x < 3) acc[threadIdx.x] = 0.0;
    __syncthreads();
    if ((threadIdx.x & 31) == 0) {
        atomicAdd(&acc[0], (double)wx);
        atomicAdd(&acc[1], (double)wy);
        atomicAdd(&acc[2], (double)wz);
    }
    __syncthreads();
    if (threadIdx.x == 0) {
        atomicAdd(&ws[4], acc[0]);
        atomicAdd(&ws[5], acc[1]);
        atomicAdd(&ws[6], acc[2]);
    }
}

__global__ void k_final(const double* __restrict__ ws, float* __restrict__ out) {
    if (blockIdx.x == 0 && threadIdx.x == 0) {
        double na        = ws[1];
        double l_conn    = ws[0] / (na + 1e-6);
        double l_support = ws[2] / (na + 1e-6);
        double l_patch   = ws[3] / (32768.0 + 1e-6);
        const double N   = 65011712.0;   // 2048 * 31 * 32 * 32 (same for x,y,z)
        double l_smooth  = (ws[4] / N + ws[5] / N + ws[6] / N) / 3.0;
        double total = 0.5 * l_conn + 0.3 * l_patch + 0.1 * l_smooth +
                       0.2 * l_support;
        out[0] = (float)total;
        out[1] = (float)l_conn;
        out[2] = (float)l_patch;
        out[3] = (float)l_smooth;
        out[4] = (float)l_support;
    }
}

extern "C" void kernel_launch(void* const* d_in, const int* in_sizes, int n_in,
                              void* d_out, int out_size, void* d_ws,
                              size_t ws_size, hipStream_t stream) {
    (void)in_sizes; (void)n_in; (void)out_size; (void)ws_size;
    const int*   structure = (const int*)d_in[0];
    const float* emb       = (const float*)d_in[1];
    double* ws  = (double*)d_ws;
    float*  out = (float*)d_out;

    k_init  <<<1,    32,  0, stream>>>(ws);
    k_struct<<<2048, 256, 0, stream>>>(structure, ws);  // 4 voxels/thread exact
    k_patch <<<8192, 256, 0, stream>>>(structure, ws);  // 4 patches/block exact
    k_smooth<<<2048, 256, 0, stream>>>(emb, ws);        // 4 rows/thread exact
    k_final <<<1,    32,  0, stream>>>(ws, out);
}